// SemanticAlignment_27358941675636
// MI455X (gfx1250) — compile-verified
//
#include <hip/hip_runtime.h>
#include <math.h>
#include <stdint.h>

typedef __attribute__((ext_vector_type(2))) float v2f;
typedef __attribute__((ext_vector_type(8))) float v8f;

#define B_  32
#define LQ_ 64
#define LV_ 128
#define QS_ 512
#define FS_ 1024
#define BN_ 512

// ---- fast tanh: prefer gfx1250 trans-unit tanh; else 1-trans-op fallback ----
#if __has_builtin(__builtin_amdgcn_tanhf)
__device__ __forceinline__ float fast_tanh(float x) { return __builtin_amdgcn_tanhf(x); }
#elif __has_builtin(__builtin_amdgcn_tanh_f32)
__device__ __forceinline__ float fast_tanh(float x) { return __builtin_amdgcn_tanh_f32(x); }
#else
__device__ __forceinline__ float fast_tanh(float x) {
  // tanh(|x|) = (1 - e^(-2|x|)) / (1 + e^(-2|x|)); exact saturation for big |x|
  const float t = __expf(-2.0f * fabsf(x));     // v_exp_f32 (trans unit)
  const float r = (1.0f - t) / (1.0f + t);
  return copysignf(r, x);
}
#endif

// ---- CDNA5 async memory->LDS DMA (ASYNCcnt-tracked, no VGPR round trip) ----
// LDS aperture keeps the byte offset in addr[31:0], so truncating the flat
// address of a __shared__ object yields the DS address the instruction wants.
__device__ __forceinline__ void async_copy_b128(const void* g, void* lds) {
  uint32_t l = (uint32_t)(uintptr_t)lds;
  asm volatile("global_load_async_to_lds_b128 %0, %1, off"
               :: "v"(l), "v"(g)
               : "memory");
}
template <int N>
__device__ __forceinline__ void wait_asynccnt() {
  asm volatile("s_wait_asynccnt %0" :: "i"(N) : "memory");
}

// ---------------------------------------------------------------------------
// C[M,N] = A[M,K] * Bm[N,K]^T  via V_WMMA_F32_16X16X4_F32, one wave per tile.
// Double-buffered LDS, filled with global_load_async_to_lds_b128 (prefetch
// chunk k+1 while WMMAs consume chunk k).  Padded rows (stride 68 floats)
// give conflict-free fragment reads.
// ---------------------------------------------------------------------------
#define TK 64
__global__ __launch_bounds__(32)
void wmma_gemm_nt(const float* __restrict__ A, const float* __restrict__ Bm,
                  float* __restrict__ C, int M, int N, int K) {
  __shared__ __align__(16) float As[2][16][TK + 4];
  __shared__ __align__(16) float Bs[2][16][TK + 4];

  const int n0   = blockIdx.x * 16;
  const int m0   = blockIdx.y * 16;
  const int lane = threadIdx.x;
  const int row  = lane & 15;   // M-row for A frag, N-col for B frag
  const int hi   = lane >> 4;   // lane half-group

  v8f acc = {};

// issue 16 async-DMA instructions moving one 16xTK A tile + B tile into LDS
#define ISSUE_CHUNK(k0, buf)                                                 \
  {                                                                          \
    _Pragma("unroll")                                                        \
    for (int i = 0; i < 8; ++i) {                                            \
      int idx = i * 32 + lane;  /* 0..255 float4 slots */                    \
      int r   = idx >> 4;                                                    \
      int c4  = idx & 15;                                                    \
      async_copy_b128(A  + (size_t)(m0 + r) * K + (k0) + c4 * 4,             \
                      &As[buf][r][c4 * 4]);                                  \
      async_copy_b128(Bm + (size_t)(n0 + r) * K + (k0) + c4 * 4,             \
                      &Bs[buf][r][c4 * 4]);                                  \
    }                                                                        \
  }

  ISSUE_CHUNK(0, 0);
  int buf = 0;
  for (int k0 = 0; k0 < K; k0 += TK) {
    const bool prefetch = (k0 + TK) < K;
    if (prefetch) ISSUE_CHUNK(k0 + TK, buf ^ 1);

    // async loads complete in order: <=16 outstanding => current chunk landed
    if (prefetch) wait_asynccnt<16>();
    else          wait_asynccnt<0>();

    // 16 chained WMMAs over this K-chunk.
    // Frag layout (32b, 16x4): VGPR v, half h -> K = kk + 2*h + v.
#pragma unroll
    for (int kk = 0; kk < TK; kk += 4) {
      const int kb = kk + 2 * hi;
      v2f a, b;
      a.x = As[buf][row][kb];  a.y = As[buf][row][kb + 1];
      b.x = Bs[buf][row][kb];  b.y = Bs[buf][row][kb + 1];
      acc = __builtin_amdgcn_wmma_f32_16x16x4_f32(
          /*neg_a=*/false, a, /*neg_b=*/false, b,
          /*c_mod=*/(short)0, acc, /*reuse_a=*/false, /*reuse_b=*/false);
    }
    buf ^= 1;
  }
#undef ISSUE_CHUNK

  // D layout: VGPR r -> M = r + 8*hi, N = lane&15
#pragma unroll
  for (int r = 0; r < 8; ++r)
    C[(size_t)(m0 + r + 8 * hi) * N + n0 + row] = acc[r];
}

// ---------------------------------------------------------------------------
// One block per (b,q): energies (tanh-dot), softmax over Lv, aligned GEMV,
// and the concat write. Uv rereads hit L2 (12MB workspace << 192MB L2).
// ---------------------------------------------------------------------------
__global__ __launch_bounds__(256)
void fused_attn(const float* __restrict__ phr, const float* __restrict__ vis,
                const float* __restrict__ bias, const float* __restrict__ wvec,
                const float* __restrict__ Wh, const float* __restrict__ Uv,
                float* __restrict__ out_sem, float* __restrict__ out_wgt,
                float* __restrict__ out_en) {
  __shared__ __align__(16) float whb[BN_];     // Wh row + bias
  __shared__ __align__(16) float eLDS[LV_];    // raw energies
  __shared__ __align__(16) float wgtLDS[LV_];  // softmax weights

  const int bq   = blockIdx.x;       // b*LQ_ + q
  const int b    = bq / LQ_;
  const int tid  = threadIdx.x;
  const int lane = tid & 31;
  const int wave = tid >> 5;

  const float* whRow = Wh + (size_t)bq * BN_;
  whb[tid]       = whRow[tid]       + bias[tid];
  whb[tid + 256] = whRow[tid + 256] + bias[tid + 256];
  __syncthreads();

  // Per-lane register copies; float4 chunk n4 = lane + 32*j (lane-coalesced).
  float4 wr[4], hr[4];
  const float4* w4 = (const float4*)wvec;
  const float4* h4 = (const float4*)whb;
#pragma unroll
  for (int j = 0; j < 4; ++j) {
    wr[j] = w4[lane + 32 * j];
    hr[j] = h4[lane + 32 * j];
  }

  // ---- energies: each wave sweeps v = wave, wave+8, ... ----
  for (int v = wave; v < LV_; v += 8) {
    const float4* uv4 = (const float4*)(Uv + ((size_t)b * LV_ + v) * BN_);
    float acc = 0.f;
#pragma unroll
    for (int j = 0; j < 4; ++j) {
      float4 u = uv4[lane + 32 * j];              // global_load_b128, coalesced
      acc += wr[j].x * fast_tanh(hr[j].x + u.x);
      acc += wr[j].y * fast_tanh(hr[j].y + u.y);
      acc += wr[j].z * fast_tanh(hr[j].z + u.z);
      acc += wr[j].w * fast_tanh(hr[j].w + u.w);
    }
#pragma unroll
    for (int off = 16; off > 0; off >>= 1) acc += __shfl_xor(acc, off, 32);
    if (lane == 0) eLDS[v] = acc;
  }
  __syncthreads();

  // ---- softmax over 128 (wave 0, 4 values/lane), write weights+energies ----
  if (wave == 0) {
    float4 e = ((float4*)eLDS)[lane];
    float m = fmaxf(fmaxf(e.x, e.y), fmaxf(e.z, e.w));
#pragma unroll
    for (int off = 16; off > 0; off >>= 1) m = fmaxf(m, __shfl_xor(m, off, 32));
    float4 ex;
    ex.x = __expf(e.x - m); ex.y = __expf(e.y - m);
    ex.z = __expf(e.z - m); ex.w = __expf(e.w - m);
    float s = ex.x + ex.y + ex.z + ex.w;
#pragma unroll
    for (int off = 16; off > 0; off >>= 1) s += __shfl_xor(s, off, 32);
    const float inv = 1.0f / s;
    ex.x *= inv; ex.y *= inv; ex.z *= inv; ex.w *= inv;
    ((float4*)wgtLDS)[lane] = ex;
    ((float4*)(out_en  + (size_t)bq * LV_))[lane] = e;
    ((float4*)(out_wgt + (size_t)bq * LV_))[lane] = ex;
  }
  __syncthreads();

  // ---- aligned[b,q,f] = sum_v w[v]*vis[b,v,f]; f4 = tid (coalesced) ----
  float4 acc4 = make_float4(0.f, 0.f, 0.f, 0.f);
  const float4* vis4 = (const float4*)(vis + (size_t)b * LV_ * FS_);
#pragma unroll 8
  for (int v = 0; v < LV_; ++v) {
    const float wv = wgtLDS[v];
    const float4 vv = vis4[(size_t)v * (FS_ / 4) + tid];
    acc4.x += wv * vv.x; acc4.y += wv * vv.y;
    acc4.z += wv * vv.z; acc4.w += wv * vv.w;
  }

  float* sem = out_sem + (size_t)bq * (QS_ + FS_);
  if (tid < 128)  // copy phr_feats into [0:512)
    ((float4*)sem)[tid] = ((const float4*)(phr + (size_t)bq * QS_))[tid];
  ((float4*)(sem + QS_))[tid] = acc4;  // aligned into [512:1536)
}

// ---------------------------------------------------------------------------
extern "C" void kernel_launch(void* const* d_in, const int* in_sizes, int n_in,
                              void* d_out, int out_size, void* d_ws, size_t ws_size,
                              hipStream_t stream) {
  (void)in_sizes; (void)n_in; (void)out_size; (void)ws_size;

  const float* phr  = (const float*)d_in[0];  // [32,64,512]
  const float* vis  = (const float*)d_in[1];  // [32,128,1024]
  const float* W    = (const float*)d_in[2];  // [512,512]
  const float* U    = (const float*)d_in[3];  // [512,1024]
  const float* bias = (const float*)d_in[4];  // [512]
  const float* w    = (const float*)d_in[5];  // [1,512]

  float* Wh = (float*)d_ws;                       // [2048,512]  4 MB
  float* Uv = Wh + (size_t)B_ * LQ_ * BN_;        // [4096,512]  8 MB

  float* out_sem = (float*)d_out;                             // [32,64,1536]
  float* out_wgt = out_sem + (size_t)B_ * LQ_ * (QS_ + FS_);  // [32,64,128]
  float* out_en  = out_wgt + (size_t)B_ * LQ_ * LV_;          // [32,64,128]

  dim3 g1(BN_ / 16, (B_ * LQ_) / 16);
  wmma_gemm_nt<<<g1, 32, 0, stream>>>(phr, W, Wh, B_ * LQ_, BN_, QS_);

  dim3 g2(BN_ / 16, (B_ * LV_) / 16);
  wmma_gemm_nt<<<g2, 32, 0, stream>>>(vis, U, Uv, B_ * LV_, BN_, FS_);

  fused_attn<<<B_ * LQ_, 256, 0, stream>>>(phr, vis, bias, w, Wh, Uv,
                                           out_sem, out_wgt, out_en);
}